// Attention_34471407518209
// MI455X (gfx1250) — compile-verified
//
#include <hip/hip_runtime.h>
#include <hip/hip_bf16.h>

#define CC 64
#define NN 16384
#define BB 32
#define EPSF 1e-5f
#define CHUNK 1024
#define NCHUNK (NN / CHUNK)   // 16

typedef float v2f __attribute__((ext_vector_type(2)));
typedef float v4f __attribute__((ext_vector_type(4)));
typedef float v8f __attribute__((ext_vector_type(8)));

// ---- workspace layout (in floats) ----
// G    : BB * 64*64   per-batch Gram matrices (atomically accumulated)
// srow : BB * 64      per-batch per-channel row sums of x
// vcol : BB * NN      per-batch channel means (v)
// M    : 64*64        Wq2 * diag(D) * Wq
// u    : 64           Wq2 @ e
// w    : BB * 64      Wc @ attnmax  (per-batch output scale vector)
static constexpr size_t OFF_G = 0;
static constexpr size_t OFF_S = OFF_G + (size_t)BB * CC * CC;   // 131072
static constexpr size_t OFF_V = OFF_S + (size_t)BB * CC;        // 133120
static constexpr size_t OFF_M = OFF_V + (size_t)BB * NN;        // 657408
static constexpr size_t OFF_U = OFF_M + (size_t)CC * CC;        // 661504
static constexpr size_t OFF_W = OFF_U + (size_t)CC;             // 661568
static constexpr size_t ZERO_FLOATS = OFF_V;                    // zero G + srow

// ------------------------------------------------------------------
// Kernel 0: zero the accumulated regions (G, srow) each launch
// ------------------------------------------------------------------
__global__ __launch_bounds__(256) void k_zero(float* __restrict__ ws) {
  size_t i = (size_t)blockIdx.x * 256 + threadIdx.x;
  if (i < ZERO_FLOATS) ws[i] = 0.0f;
}

// ------------------------------------------------------------------
// Kernel 1: per-batch Gram G[b] = x[b] x[b]^T via V_WMMA_F32_16X16X4_F32,
//           plus row sums s[b] and channel means v[b,n].
// grid = (NCHUNK, BB), block = 256 (8 waves). Each wave owns 128 columns.
// ------------------------------------------------------------------
__global__ __launch_bounds__(256) void k_gram(const float* __restrict__ x,
                                              float* __restrict__ G,
                                              float* __restrict__ srow,
                                              float* __restrict__ vcol) {
  const int b = blockIdx.y;
  const int chunk0 = blockIdx.x * CHUNK;
  const int tid = threadIdx.x;
  const int lane = tid & 31;
  const int wave = tid >> 5;
  const float* xb = x + (size_t)b * CC * NN;

  __shared__ float gsm[CC * CC];
  for (int i = tid; i < CC * CC; i += 256) gsm[i] = 0.0f;
  __syncthreads();

  // ---- WMMA Gram accumulation over this wave's 128 columns ----
  v8f acc[4][4] = {};
  const int half = lane >> 4;    // which lane-half (maps to K pairs)
  const int lm   = lane & 15;    // M / N index within tile
  const int n_lo = chunk0 + wave * (CHUNK / 8);
  const int n_hi = n_lo + (CHUNK / 8);

  // Fragment layout for f32 16x16x4:
  //   A[m,k] lives at lane = (k>=2)*16 + m, vgpr = k&1
  //   B[k,n] lives at lane = (k>=2)*16 + n, vgpr = k&1
  // Both map to x[cb*16 + lm, n0 + (k&1) + 2*half] -> one float2 load
  // serves as the A- and B-fragment of channel block cb.
#pragma unroll 1
  for (int n0 = n_lo; n0 < n_hi; n0 += 4) {
    v2f f[4];
#pragma unroll
    for (int cb = 0; cb < 4; ++cb) {
      const float* p = xb + (size_t)(cb * 16 + lm) * NN + (n0 + 2 * half);
      f[cb] = *(const v2f*)p;                      // 8B aligned (n0 multiple of 4)
    }
#pragma unroll
    for (int i = 0; i < 4; ++i)
#pragma unroll
      for (int j = 0; j < 4; ++j)
        acc[i][j] = __builtin_amdgcn_wmma_f32_16x16x4_f32(
            false, f[i], false, f[j], (short)0, acc[i][j], false, false);
  }

  // ---- reduce 8 waves' tiles into LDS, then one atomic sweep to global ----
  // C/D tile layout: lanes 0-15 -> N=lane, M=r ; lanes 16-31 -> N=lane-16, M=r+8
#pragma unroll
  for (int i = 0; i < 4; ++i)
#pragma unroll
    for (int j = 0; j < 4; ++j)
#pragma unroll
      for (int r = 0; r < 8; ++r) {
        int gm = i * 16 + r + 8 * half;
        int gn = j * 16 + lm;
        atomicAdd(&gsm[gm * CC + gn], acc[i][j][r]);
      }
  __syncthreads();
  for (int i = tid; i < CC * CC; i += 256)
    atomicAdd(&G[(size_t)b * CC * CC + i], gsm[i]);

  // ---- channel means v[b,n] (each column owned by exactly one thread) ----
#pragma unroll 1
  for (int i = 0; i < 4; ++i) {
    int n = chunk0 + i * 256 + tid;
    float sum = 0.0f;
#pragma unroll 1
    for (int c = 0; c < CC; ++c) sum += xb[(size_t)c * NN + n];
    vcol[(size_t)b * NN + n] = sum * (1.0f / CC);
  }

  // ---- row sums s[b,c] ----
  {
    int r = tid & 63;
    int grp = tid >> 6;                            // 4 groups x 256 cols = CHUNK
    const float* rp = xb + (size_t)r * NN + chunk0 + grp * 256;
    float sum = 0.0f;
#pragma unroll 1
    for (int n = 0; n < 256; ++n) sum += rp[n];
    atomicAdd(&srow[b * CC + r], sum);
  }
}

// ------------------------------------------------------------------
// Kernel 2: BatchNorm statistics from Gbar = sum_b G[b], sbar = sum_b s[b];
//           M = Wq2 * diag(gamma/sigma) * Wq ; u = Wq2 @ e.   (1 block)
// ------------------------------------------------------------------
__global__ __launch_bounds__(256) void k_stats(const float* __restrict__ G,
                                               const float* __restrict__ srow,
                                               const float* __restrict__ Wq,
                                               const float* __restrict__ bq,
                                               const float* __restrict__ gamma,
                                               const float* __restrict__ beta,
                                               const float* __restrict__ Wq2,
                                               float* __restrict__ M,
                                               float* __restrict__ u) {
  __shared__ float Gbar[CC * CC];
  __shared__ float sbar[CC];
  __shared__ float Dv[CC];
  __shared__ float ev[CC];
  const int tid = threadIdx.x;

  for (int i = tid; i < CC * CC; i += 256) {
    float sum = 0.0f;
    for (int b2 = 0; b2 < BB; ++b2) sum += G[(size_t)b2 * CC * CC + i];
    Gbar[i] = sum;
  }
  if (tid < CC) {
    float sum = 0.0f;
    for (int b2 = 0; b2 < BB; ++b2) sum += srow[b2 * CC + tid];
    sbar[tid] = sum;
  }
  __syncthreads();

  const float inv = 1.0f / ((float)BB * (float)NN);
  if (tid < CC) {
    const int c = tid;
    const float* wr = Wq + c * CC;
    float dotm = 0.0f;
    for (int d = 0; d < CC; ++d) dotm += wr[d] * sbar[d];
    float mu = dotm * inv + bq[c];
    float quad = 0.0f;
    for (int d = 0; d < CC; ++d) {
      float t = 0.0f;
      for (int e2 = 0; e2 < CC; ++e2) t += Gbar[d * CC + e2] * wr[e2];
      quad += wr[d] * t;
    }
    // E[q^2] = wGw/(BN) + 2*bq*w.xbar + bq^2 ; var = E[q^2] - mu^2
    float m2 = quad * inv + 2.0f * bq[c] * dotm * inv + bq[c] * bq[c];
    float var = m2 - mu * mu;
    float Dc = gamma[c] * rsqrtf(var + EPSF);
    Dv[c] = Dc;
    ev[c] = Dc * (bq[c] - mu) + beta[c];
  }
  __syncthreads();

  for (int i = tid; i < CC * CC; i += 256) {
    int o = i >> 6, d = i & 63;
    float sum = 0.0f;
    for (int c = 0; c < CC; ++c) sum += Wq2[o * CC + c] * Dv[c] * Wq[c * CC + d];
    M[i] = sum;
  }
  if (tid < CC) {
    float sum = 0.0f;
    for (int c = 0; c < CC; ++c) sum += Wq2[tid * CC + c] * ev[c];
    u[tid] = sum;
  }
}

// ------------------------------------------------------------------
// Kernel 3: per batch: attn = M*G[b]*Wsr^T + (u+bq2) y^T + (z + N(u+bq2)) bsr^T
//           with y = Wsr s[b], z = M s[b]; then rowmax, w[b] = Wc @ rowmax.
// grid = BB, block = 256
// ------------------------------------------------------------------
__global__ __launch_bounds__(256) void k_attn(const float* __restrict__ G,
                                              const float* __restrict__ srow,
                                              const float* __restrict__ M,
                                              const float* __restrict__ u,
                                              const float* __restrict__ Wsr,
                                              const float* __restrict__ bsr,
                                              const float* __restrict__ bq2,
                                              const float* __restrict__ Wc,
                                              float* __restrict__ w) {
  const int b = blockIdx.x;
  const int tid = threadIdx.x;
  __shared__ float Gs[CC * CC];
  __shared__ float Ms[CC * CC];
  __shared__ float t1[CC * CC];
  __shared__ float yv[CC], zv[CC], maxp[CC * 4], rmax[CC];

  for (int i = tid; i < CC * CC; i += 256) {
    Gs[i] = G[(size_t)b * CC * CC + i];
    Ms[i] = M[i];
  }
  __syncthreads();

  if (tid < CC) {
    float ys = 0.0f, zs = 0.0f;
    for (int d = 0; d < CC; ++d) {
      float sd = srow[b * CC + d];
      ys += Wsr[tid * CC + d] * sd;
      zs += Ms[tid * CC + d] * sd;
    }
    yv[tid] = ys;
    zv[tid] = zs;
  }

  for (int i = tid; i < CC * CC; i += 256) {       // t1 = M @ G[b]
    int o = i >> 6, d = i & 63;
    float sum = 0.0f;
    for (int c = 0; c < CC; ++c) sum += Ms[o * CC + c] * Gs[c * CC + d];
    t1[i] = sum;
  }
  __syncthreads();

  {
    int o = tid >> 2, pg = tid & 3;                // 4 threads per row
    float ub = u[o] + bq2[o];
    float zb = zv[o] + (float)NN * ub;
    float mx = -3.402823466e38f;
    for (int pp = 0; pp < 16; ++pp) {
      int p = pg * 16 + pp;
      float sum = 0.0f;
      for (int d = 0; d < CC; ++d) sum += t1[o * CC + d] * Wsr[p * CC + d];
      float a = sum + ub * yv[p] + zb * bsr[p];
      mx = fmaxf(mx, a);
    }
    maxp[o * 4 + pg] = mx;
  }
  __syncthreads();
  if (tid < CC) {
    float mx = fmaxf(fmaxf(maxp[tid * 4], maxp[tid * 4 + 1]),
                     fmaxf(maxp[tid * 4 + 2], maxp[tid * 4 + 3]));
    rmax[tid] = mx;
  }
  __syncthreads();
  if (tid < CC) {
    float sum = 0.0f;
    for (int c = 0; c < CC; ++c) sum += Wc[tid * CC + c] * rmax[c];
    w[b * CC + tid] = sum;
  }
}

// ------------------------------------------------------------------
// Kernel 4: out[b,o,n] = w[b,o] * v[b,n]  (streaming rank-1, NT stores)
// grid = (BB*CC, NN/1024), block = 256, float4 per thread
// ------------------------------------------------------------------
__global__ __launch_bounds__(256) void k_out(const float* __restrict__ w,
                                             const float* __restrict__ vcol,
                                             float* __restrict__ out) {
  const int bo = blockIdx.x;                       // b*64 + o
  const int b = bo >> 6;
  const int n4 = (blockIdx.y * 256 + threadIdx.x) * 4;
  const float s = w[bo];
  const v4f vv = *(const v4f*)(vcol + (size_t)b * NN + n4);
  v4f r = vv * s;
  __builtin_nontemporal_store(r, (v4f*)(out + (size_t)bo * NN + n4));
}

// ------------------------------------------------------------------
extern "C" void kernel_launch(void* const* d_in, const int* in_sizes, int n_in,
                              void* d_out, int out_size, void* d_ws, size_t ws_size,
                              hipStream_t stream) {
  (void)in_sizes; (void)n_in; (void)out_size; (void)ws_size;
  const float* x     = (const float*)d_in[0];
  const float* Wq    = (const float*)d_in[1];
  const float* bq    = (const float*)d_in[2];
  const float* gamma = (const float*)d_in[3];
  const float* beta  = (const float*)d_in[4];
  const float* Wq2   = (const float*)d_in[5];
  const float* bq2   = (const float*)d_in[6];
  const float* Wsr   = (const float*)d_in[7];
  const float* bsr   = (const float*)d_in[8];
  const float* Wc    = (const float*)d_in[9];

  float* ws   = (float*)d_ws;
  float* G    = ws + OFF_G;
  float* srow = ws + OFF_S;
  float* vcol = ws + OFF_V;
  float* M    = ws + OFF_M;
  float* u    = ws + OFF_U;
  float* w    = ws + OFF_W;

  k_zero<<<(int)((ZERO_FLOATS + 255) / 256), 256, 0, stream>>>(ws);
  k_gram<<<dim3(NCHUNK, BB), 256, 0, stream>>>(x, G, srow, vcol);
  k_stats<<<1, 256, 0, stream>>>(G, srow, Wq, bq, gamma, beta, Wq2, M, u);
  k_attn<<<BB, 256, 0, stream>>>(G, srow, M, u, Wsr, bsr, bq2, Wc, w);
  k_out<<<dim3(BB * CC, NN / 1024), 256, 0, stream>>>(w, vcol, (float*)d_out);
}